// MLItemLoading_83743272337681
// MI455X (gfx1250) — compile-verified
//
#include <hip/hip_runtime.h>
#include <math.h>

#define BATCH 16384
#define XCOLS 10242
#define KC    128          // K columns staged per chunk
#define WAVES 4            // waves (16-row tiles) per workgroup

typedef __attribute__((ext_vector_type(2))) float v2f;
typedef __attribute__((ext_vector_type(8))) float v8f;

// out layout: (B, 128) = [rate 0:32 | genre 32:64 | director 64:96 | actor 96:128]

__global__ __launch_bounds__(128) void pool_wmma_kernel(
    const int* __restrict__ x,
    const float* __restrict__ Wg, const float* __restrict__ Wd,
    const float* __restrict__ Wa, float* __restrict__ out)
{
    __shared__ float ldsX[WAVES][16 * KC];   // A fragments, swizzled, per wave
    __shared__ float ldsW[2][16 * KC];       // B fragments, swizzled, shared (2 N-tiles)
    __shared__ float ldsP[WAVES][32];        // row-sum partials

    const int t  = threadIdx.x;
    const int wv = t >> 5;
    const int l  = t & 31;
    const int rowBase = (blockIdx.x * WAVES + wv) * 16;

    const float* Ws[3]     = { Wg, Wd, Wa };
    const int    colS[3]   = { 1, 26, 2212 };
    const int    Ksz[3]    = { 25, 2186, 8030 };
    const int    outB[3]   = { 32, 64, 96 };

    for (int seg = 0; seg < 3; ++seg) {
        const float* W = Ws[seg];
        const int K  = Ksz[seg];
        const int cs = colS[seg];
        v8f acc0 = {}; v8f acc1 = {};
        float rsum = 0.0f;
        const int nChunks = (K + KC - 1) / KC;

        for (int c = 0; c < nChunks; ++c) {
            const int k0 = c * KC;
            __syncthreads();   // LDS WAR vs previous compute phase

            // ---- stage X: wave stages its own 16 rows; 128B-contiguous loads,
            //      cvt int->f32, scatter into A-fragment (16x4 f32) swizzle order
            for (int rr = 0; rr < 16; ++rr) {
                const long rowOff = (long)(rowBase + rr) * XCOLS + cs + k0;
                #pragma unroll
                for (int ph = 0; ph < 4; ++ph) {
                    const int k = ph * 32 + l;             // 0..127 rel chunk
                    const float f = (k0 + k < K) ? (float)x[rowOff + k] : 0.0f;
                    const int s = k >> 2, r = k & 3;
                    ldsX[wv][(s * 32 + rr + 16 * (r >> 1)) * 2 + (r & 1)] = f;
                }
            }
            // ---- stage W: whole block; each iter loads 128 contiguous dwords of
            //      one emb-row e, scattered into B-fragment (4x16 f32) swizzle order
            for (int e = 0; e < 32; ++e) {
                const int k = t;                            // 0..127 rel chunk
                const float f = (k0 + k < K) ? W[(long)e * K + k0 + k] : 0.0f;
                const int s = k >> 2, r = k & 3;
                ldsW[e >> 4][(s * 32 + (e & 15) + 16 * (r >> 1)) * 2 + (r & 1)] = f;
            }
            __syncthreads();

            // ---- compute: D = A(16x4) x B(4x16) + C, fp32 end-to-end
            #pragma unroll 4
            for (int s = 0; s < KC / 4; ++s) {
                v2f a  = *(const v2f*)&ldsX[wv][s * 64 + l * 2];
                v2f b0 = *(const v2f*)&ldsW[0][s * 64 + l * 2];
                v2f b1 = *(const v2f*)&ldsW[1][s * 64 + l * 2];
                acc0 = __builtin_amdgcn_wmma_f32_16x16x4_f32(
                           false, a, false, b0, (short)0, acc0, false, false);
                acc1 = __builtin_amdgcn_wmma_f32_16x16x4_f32(
                           false, a, false, b1, (short)0, acc1, false, false);
                rsum += a[0] + a[1];   // row-sum partial (K%4 in {0,1} or {2,3})
            }
        }

        // ---- row sums: lane l holds half of row (l&15)'s sum; combine via LDS
        ldsP[wv][l] = rsum;
        __syncthreads();

        // ---- epilogue: C/D layout VGPR j -> row m = j + 8*(l>>4), col n = l&15
        #pragma unroll
        for (int j = 0; j < 8; ++j) {
            const int m = j + ((l >> 4) << 3);
            const float sum = ldsP[wv][m] + ldsP[wv][m + 16];
            const long ob = (long)(rowBase + m) * 128 + outB[seg];
            out[ob + (l & 15)]      = acc0[j] / sum;
            out[ob + 16 + (l & 15)] = acc1[j] / sum;
        }
        __syncthreads();
    }
}

// max-norm gather: one wave32 per batch row, lane = emb dim
__global__ __launch_bounds__(256) void rate_kernel(
    const int* __restrict__ x, const float* __restrict__ table,
    float* __restrict__ out)
{
    const int b = blockIdx.x * 8 + (threadIdx.x >> 5);
    const int e = threadIdx.x & 31;
    if (b >= BATCH) return;                       // wave-uniform
    const int idx = x[(long)b * XCOLS];
    const float v = table[idx * 32 + e];
    float s = v * v;
    // wave32 butterfly reduction via ds_swizzle (xor 16,8,4,2,1)
    s += __int_as_float(__builtin_amdgcn_ds_swizzle(__float_as_int(s), (16 << 10) | 0x1F));
    s += __int_as_float(__builtin_amdgcn_ds_swizzle(__float_as_int(s), ( 8 << 10) | 0x1F));
    s += __int_as_float(__builtin_amdgcn_ds_swizzle(__float_as_int(s), ( 4 << 10) | 0x1F));
    s += __int_as_float(__builtin_amdgcn_ds_swizzle(__float_as_int(s), ( 2 << 10) | 0x1F));
    s += __int_as_float(__builtin_amdgcn_ds_swizzle(__float_as_int(s), ( 1 << 10) | 0x1F));
    const float norm  = sqrtf(s);
    const float scale = (norm > 1.0f) ? (1.0f / (norm + 1e-7f)) : 1.0f;
    out[(long)b * 128 + e] = v * scale;
}

extern "C" void kernel_launch(void* const* d_in, const int* in_sizes, int n_in,
                              void* d_out, int out_size, void* d_ws, size_t ws_size,
                              hipStream_t stream) {
    (void)in_sizes; (void)n_in; (void)out_size; (void)d_ws; (void)ws_size;
    const int*   xin = (const int*)d_in[0];
    const float* rt  = (const float*)d_in[1];
    const float* Wg  = (const float*)d_in[2];
    const float* Wd  = (const float*)d_in[3];
    const float* Wa  = (const float*)d_in[4];
    float* out = (float*)d_out;

    pool_wmma_kernel<<<BATCH / (16 * WAVES), 128, 0, stream>>>(xin, Wg, Wd, Wa, out);
    rate_kernel<<<BATCH / 8, 256, 0, stream>>>(xin, rt, out);
}